// GRU_15547781611826
// MI455X (gfx1250) — compile-verified
//
#include <hip/hip_runtime.h>
#include <hip/hip_bf16.h>
#include <math.h>

// ---------------------------------------------------------------------------
// GRU forward + Jacobians for MI455X (gfx1250, wave32).
// B=32, S=512, I=H=128, O=32.  Output = [y (32x32) | jcbT (512x32x128x128)].
// Bandwidth-bound on the 1 GiB Jacobian write; fp32 WMMA for all GEMMs.
// ---------------------------------------------------------------------------

typedef float v2f __attribute__((ext_vector_type(2)));
typedef float v8f __attribute__((ext_vector_type(8)));
typedef float f4  __attribute__((ext_vector_type(4)));

static __device__ __forceinline__ v8f wmma_f32_4(v2f a, v2f b, v8f c) {
  // V_WMMA_F32_16X16X4_F32 : D = A(16x4) * B(4x16) + C(16x16), fp32
  return __builtin_amdgcn_wmma_f32_16x16x4_f32(
      /*neg_a=*/false, a, /*neg_b=*/false, b,
      /*c_mod=*/(short)0, c, /*reuse_a=*/false, /*reuse_b=*/false);
}

static __device__ __forceinline__ float sigmoidf(float x) {
  return 1.0f / (1.0f + expf(-x));
}

#define S_LEN 512
#define BATCH 32
#define HID   128
#define G3    384   // 3*HID

// ---------------------------------------------------------------------------
// K0: transpose weights into workspace.
//   whhT[blk][k][j] = W_hh[blk*128 + j][k]   (blk 0:r, 1:z, 2:n) -> 49152 f
//   wihT[k][g]      = W_ih[g][k]                                  -> 49152 f
// ---------------------------------------------------------------------------
__global__ void gru_k0_transpose(const float* __restrict__ W_hh,
                                 const float* __restrict__ W_ih,
                                 float* __restrict__ whhT,
                                 float* __restrict__ wihT) {
  int idx = blockIdx.x * 256 + threadIdx.x;
  if (idx < 3 * HID * HID) {
    int blk = idx >> 14, rem = idx & 16383;
    int k = rem >> 7, j = rem & 127;
    whhT[idx] = W_hh[((blk << 7) + j) * HID + k];
  } else if (idx < 2 * 3 * HID * HID) {
    int e = idx - 3 * HID * HID;
    int k = e / G3, g = e - k * G3;
    wihT[(size_t)k * G3 + g] = W_ih[g * HID + k];
  }
}

// ---------------------------------------------------------------------------
// K1: ih[row][g] = sum_k x[b][s][k] * W_ih[g][k] + b_ih[g], row = s*32+b.
// 1024 WGs x 256 threads; one M-tile (16 rows) per WG, 8 waves x 3 N-tiles.
// ---------------------------------------------------------------------------
__global__ void gru_k1_ih(const float* __restrict__ x,
                          const float* __restrict__ wihT,
                          const float* __restrict__ b_ih,
                          float* __restrict__ ih) {
  __shared__ float a_s[16 * 132];  // padded stride to dodge LDS bank conflicts
  const int tid = threadIdx.x;
  const int rowbase = blockIdx.x * 16;

  for (int i = tid; i < 16 * HID; i += 256) {
    int r = i >> 7, k = i & 127;
    int row = rowbase + r;
    int s = row >> 5, b = row & 31;
    a_s[r * 132 + k] = x[((size_t)b * S_LEN + s) * HID + k];
  }
  __syncthreads();

  const int wave = tid >> 5, lane = tid & 31;
  const int mrow = lane & 15;
  const int kk = (lane >> 4) << 1;

  for (int nt = wave * 3; nt < wave * 3 + 3; ++nt) {
    v8f acc = {};
    const int col = nt * 16 + mrow;
    for (int k0 = 0; k0 < HID; k0 += 4) {
      v2f a, b;
      a.x = a_s[mrow * 132 + k0 + kk];
      a.y = a_s[mrow * 132 + k0 + kk + 1];
      b.x = wihT[(size_t)(k0 + kk) * G3 + col];
      b.y = wihT[(size_t)(k0 + kk + 1) * G3 + col];
      acc = wmma_f32_4(a, b, acc);
    }
    const float bias = b_ih[col];
    for (int m = 0; m < 8; ++m) {
      int r = m + ((lane >> 4) << 3);
      ih[(size_t)(rowbase + r) * G3 + col] = acc[m] + bias;
    }
  }
}

// ---------------------------------------------------------------------------
// K2: sequential scan. 1 WG of 256 threads (8 waves).
// LDS: h (32x132), hh (32x384), W_hh^T fully resident (49152 f) -> ~257 KB
// (CDNA5 allows up to 320 KB LDS per workgroup).
// ---------------------------------------------------------------------------
__global__ void gru_k2_scan(const float* __restrict__ ih,
                            const float* __restrict__ whhT,
                            const float* __restrict__ b_hh,
                            const float* __restrict__ W_y,
                            const float* __restrict__ b_y,
                            float* __restrict__ h_prev,
                            float* __restrict__ y) {
  extern __shared__ float smem[];
  float* h_s  = smem;                    // 32*132 = 4224
  float* hh_s = h_s + 32 * 132;          // 32*384 = 12288
  float* w_s  = hh_s + 32 * G3;          // 49152, layout blk*16384 + k*128 + j

  const int tid = threadIdx.x;
  for (int i = tid; i < 3 * HID * HID; i += 256) w_s[i] = whhT[i];
  for (int i = tid; i < 32 * 132; i += 256) h_s[i] = 0.0f;
  __syncthreads();

  const int wave = tid >> 5, lane = tid & 31;
  const int mrow = lane & 15;
  const int kk = (lane >> 4) << 1;

  for (int s = 0; s < S_LEN; ++s) {
    // h_prev[s] = h (state entering step s)
    for (int i = tid; i < BATCH * HID; i += 256) {
      int b = i >> 7, j = i & 127;
      h_prev[(size_t)s * (BATCH * HID) + i] = h_s[b * 132 + j];
    }

    // hh = h @ W_hh^T + b_hh : 2 M-tiles x 24 N-tiles = 48 tiles / 8 waves
    for (int t = wave * 6; t < wave * 6 + 6; ++t) {
      int mt = t / 24, nt = t - mt * 24;
      v8f acc = {};
      const int col = nt * 16 + mrow;
      const int widx = ((col >> 7) << 14) + (col & 127);
      for (int k0 = 0; k0 < HID; k0 += 4) {
        v2f a, b;
        a.x = h_s[(mt * 16 + mrow) * 132 + k0 + kk];
        a.y = h_s[(mt * 16 + mrow) * 132 + k0 + kk + 1];
        b.x = w_s[widx + (k0 + kk) * HID];
        b.y = w_s[widx + (k0 + kk + 1) * HID];
        acc = wmma_f32_4(a, b, acc);
      }
      const float bias = b_hh[col];
      for (int m = 0; m < 8; ++m) {
        int r = mt * 16 + m + ((lane >> 4) << 3);
        hh_s[r * G3 + col] = acc[m] + bias;
      }
    }
    __syncthreads();

    // gate elementwise + state update
    for (int i = tid; i < BATCH * HID; i += 256) {
      int b = i >> 7, j = i & 127;
      size_t base = (size_t)(s * BATCH + b) * G3;
      float hr = hh_s[b * G3 + j];
      float hz = hh_s[b * G3 + HID + j];
      float hn = hh_s[b * G3 + 2 * HID + j];
      float r_ = sigmoidf(ih[base + j] + hr);
      float z_ = sigmoidf(ih[base + HID + j] + hz);
      float n_ = tanhf(ih[base + 2 * HID + j] + r_ * hn);
      float hold = h_s[b * 132 + j];
      h_s[b * 132 + j] = (1.0f - z_) * n_ + z_ * hold;
    }
    __syncthreads();
  }

  // y = h_last @ W_y^T + b_y  (32x32, 4 WMMA tiles, waves 0..3)
  if (wave < 4) {
    int mt = wave >> 1, nt = wave & 1;
    v8f acc = {};
    const int col = nt * 16 + mrow;
    for (int k0 = 0; k0 < HID; k0 += 4) {
      v2f a, b;
      a.x = h_s[(mt * 16 + mrow) * 132 + k0 + kk];
      a.y = h_s[(mt * 16 + mrow) * 132 + k0 + kk + 1];
      b.x = W_y[col * HID + k0 + kk];
      b.y = W_y[col * HID + k0 + kk + 1];
      acc = wmma_f32_4(a, b, acc);
    }
    const float bias = b_y[col];
    for (int m = 0; m < 8; ++m) {
      int r = mt * 16 + m + ((lane >> 4) << 3);
      y[r * 32 + col] = acc[m] + bias;
    }
  }
}

// ---------------------------------------------------------------------------
// K3: Jacobians. 1024 WGs x 256 threads; 16 (s,b) rows per WG.
//   hh = h_prev @ W_hh^T + b_hh  (WMMA), then coefficient vectors in LDS,
//   then the 1 GiB broadcast:
//   out[511-s][b][k][j] = WrT[k][j]*cr[j] + WnT[k][j]*cn[j]
//                       + WzT[k][j]*cz[j] + (k==j)*z[j]
//   written with nontemporal float4 stores (streaming output).
// ---------------------------------------------------------------------------
__global__ void gru_k3_jac(const float* __restrict__ ih,
                           const float* __restrict__ h_prev,
                           const float* __restrict__ whhT,
                           const float* __restrict__ b_hh,
                           float* __restrict__ jac) {
  extern __shared__ float smem3[];
  float* h_s  = smem3;               // 16*132 = 2112
  float* hh_s = h_s + 16 * 132;      // 16*384 = 6144
  float* cr_s = hh_s + 16 * G3;      // 2048
  float* cn_s = cr_s + 2048;         // 2048
  float* cz_s = cn_s + 2048;         // 2048
  float* zz_s = cz_s + 2048;         // 2048   -> total 16448 f = 65792 B

  const int tid = threadIdx.x;
  const int rowbase = blockIdx.x * 16;

  for (int i = tid; i < 16 * HID; i += 256) {
    int r = i >> 7, k = i & 127;
    h_s[r * 132 + k] = h_prev[(size_t)(rowbase + r) * HID + k];
  }
  __syncthreads();

  const int wave = tid >> 5, lane = tid & 31;
  const int mrow = lane & 15;
  const int kk = (lane >> 4) << 1;

  // hh = h_prev @ W_hh^T + b_hh : 1 M-tile x 24 N-tiles / 8 waves
  for (int nt = wave * 3; nt < wave * 3 + 3; ++nt) {
    v8f acc = {};
    const int col = nt * 16 + mrow;
    const int widx = ((col >> 7) << 14) + (col & 127);
    for (int k0 = 0; k0 < HID; k0 += 4) {
      v2f a, b;
      a.x = h_s[mrow * 132 + k0 + kk];
      a.y = h_s[mrow * 132 + k0 + kk + 1];
      b.x = whhT[widx + (k0 + kk) * HID];
      b.y = whhT[widx + (k0 + kk + 1) * HID];
      acc = wmma_f32_4(a, b, acc);
    }
    const float bias = b_hh[col];
    for (int m = 0; m < 8; ++m) {
      int r = m + ((lane >> 4) << 3);
      hh_s[r * G3 + col] = acc[m] + bias;
    }
  }
  __syncthreads();

  // coefficient vectors
  for (int i = tid; i < 16 * HID; i += 256) {
    int r = i >> 7, j = i & 127;
    size_t base = (size_t)(rowbase + r) * G3;
    float hr = hh_s[r * G3 + j];
    float hz = hh_s[r * G3 + HID + j];
    float M  = hh_s[r * G3 + 2 * HID + j];
    float rr = sigmoidf(ih[base + j] + hr);
    float zz = sigmoidf(ih[base + HID + j] + hz);
    float nn = tanhf(ih[base + 2 * HID + j] + rr * M);
    float c67 = (1.0f - nn * nn) * (1.0f - zz);
    cr_s[i] = rr * (1.0f - rr) * M * c67;
    cn_s[i] = rr * c67;
    cz_s[i] = zz * (1.0f - zz) * (h_s[r * 132 + j] - nn);
    zz_s[i] = zz;
  }
  __syncthreads();

  // streaming broadcast write (s-flip folded into address)
  const int s  = rowbase >> 5;   // same s for all 16 rows in this WG
  const int b0 = rowbase & 31;
  const int jbase = lane << 2;   // this lane's 4 consecutive j values

  for (int rk = wave; rk < 16 * HID; rk += 8) {
    int r = rk >> 7, k = rk & 127;
    const f4 w0 = ((const f4*)(whhT + (size_t)k * HID))[lane];             // WrT row k
    const f4 w1 = ((const f4*)(whhT + 16384 + (size_t)k * HID))[lane];     // WzT row k
    const f4 w2 = ((const f4*)(whhT + 32768 + (size_t)k * HID))[lane];     // WnT row k
    const f4 a0 = ((const f4*)(cr_s + r * HID))[lane];
    const f4 a1 = ((const f4*)(cz_s + r * HID))[lane];
    const f4 a2 = ((const f4*)(cn_s + r * HID))[lane];
    f4 o = w0 * a0 + w2 * a2 + w1 * a1;
    if (k >= jbase && k < jbase + 4) {
      const f4 az = ((const f4*)(zz_s + r * HID))[lane];
      int d = k - jbase;
      o[d] += az[d];
    }
    size_t orow = ((size_t)(S_LEN - 1 - s) * BATCH + (b0 + r)) * (size_t)(HID * HID)
                + (size_t)k * HID;
    __builtin_nontemporal_store(o, (f4*)(jac + orow) + lane);
  }
}

// ---------------------------------------------------------------------------
// launch
// ---------------------------------------------------------------------------
extern "C" void kernel_launch(void* const* d_in, const int* in_sizes, int n_in,
                              void* d_out, int out_size, void* d_ws, size_t ws_size,
                              hipStream_t stream) {
  const float* x    = (const float*)d_in[0];  // (B,S,I)
  const float* W_ih = (const float*)d_in[1];  // (384,128)
  const float* W_hh = (const float*)d_in[2];  // (384,128)
  const float* b_ih = (const float*)d_in[3];  // (384)
  const float* b_hh = (const float*)d_in[4];  // (384)
  const float* W_y  = (const float*)d_in[5];  // (32,128)
  const float* b_y  = (const float*)d_in[6];  // (32)

  float* y   = (float*)d_out;       // 1024 floats
  float* jac = y + 32 * 32;         // 512*32*128*128 floats

  float* ws    = (float*)d_ws;
  float* ih    = ws;                          // 16384*384 = 6,291,456 f
  float* hprev = ws + 6291456;                //   512*4096 = 2,097,152 f
  float* whhT  = ws + 8388608;                //   49,152 f
  float* wihT  = ws + 8437760;                //   49,152 f

  gru_k0_transpose<<<dim3(384), dim3(256), 0, stream>>>(W_hh, W_ih, whhT, wihT);
  gru_k1_ih<<<dim3(1024), dim3(256), 0, stream>>>(x, wihT, b_ih, ih);
  gru_k2_scan<<<dim3(1), dim3(256), 262656, stream>>>(ih, whhT, b_hh, W_y, b_y,
                                                      hprev, y);
  gru_k3_jac<<<dim3(1024), dim3(256), 65792, stream>>>(ih, hprev, whhT, b_hh, jac);
}